// rela_graph_node_update_2302102471104
// MI455X (gfx1250) — compile-verified
//
#include <hip/hip_runtime.h>

#define NPROT 100000
#define NDRUG 20000
#define DIMF  128
#define EPPI  1600000
#define EDTI  800000
#define AS_STRIDE 132   // padded LDS row stride (floats): banks spread 4 apart

typedef __attribute__((ext_vector_type(2))) float v2f;
typedef __attribute__((ext_vector_type(8))) float v8f;

// ---------------------------------------------------------------------------
// degree / count helpers
// ---------------------------------------------------------------------------
__global__ void init_deg_cnt(float* __restrict__ deg, float* __restrict__ cnt, int n) {
    int i = blockIdx.x * blockDim.x + threadIdx.x;
    if (i < n) { deg[i] = 1.0f; cnt[i] = 0.0f; }   // deg starts at 1 (self-loop)
}

__global__ void count_edges(const int* __restrict__ idx, float* __restrict__ acc, int e) {
    int i = blockIdx.x * blockDim.x + threadIdx.x;
    if (i < e) atomicAdd(&acc[idx[i]], 1.0f);
}

__global__ void to_dinv(float* __restrict__ deg, int n) {
    int i = blockIdx.x * blockDim.x + threadIdx.x;
    if (i < n) deg[i] = 1.0f / sqrtf(deg[i]);      // deg >= 1 always
}

// ---------------------------------------------------------------------------
// Wt[n,k] = W[k,n]  (128x128, one-shot; makes B fragments contiguous b64)
// ---------------------------------------------------------------------------
__global__ __launch_bounds__(128) void transpose128(const float* __restrict__ W,
                                                    float* __restrict__ Wt) {
    const int n = blockIdx.x;     // 0..127
    const int k = threadIdx.x;    // 0..127
    Wt[n * DIMF + k] = W[k * DIMF + n];
}

// ---------------------------------------------------------------------------
// C[M,128] = A[M,128] @ W[128,128] + bias1 + bias2 + cadd, W given TRANSPOSED.
// One block = one 16-row M tile (staged in padded LDS); 8 waves, wave w owns
// N-tile w. 32 x V_WMMA_F32_16X16X4_F32 covers K=128 at full f32 precision.
// ---------------------------------------------------------------------------
__global__ __launch_bounds__(256) void gemm_wmma_f32(
    const float* __restrict__ A, const float* __restrict__ Wt,
    const float* __restrict__ bias1, const float* __restrict__ bias2,
    float cadd, float* __restrict__ C)
{
    __shared__ __align__(16) float As[16 * AS_STRIDE];

    const int tid  = threadIdx.x;
    const int wave = tid >> 5;
    const int lane = tid & 31;
    const int half = lane >> 4;   // lanes 16-31 hold the "upper" K slice
    const int l16  = lane & 15;
    const int m0   = blockIdx.x * 16;
    const int n0   = wave * 16;

    // Stage the 16x128 A tile to LDS with padded rows (conflict-free reads).
    const float4* A4 = (const float4*)(A + (size_t)m0 * DIMF);
    #pragma unroll
    for (int i = 0; i < 2; ++i) {
        const int idx = tid + i * 256;        // 0..511 float4s
        const int r = idx >> 5;               // tile row
        const int c = idx & 31;               // float4 within row
        ((float4*)(As + r * AS_STRIDE))[c] = A4[idx];
    }
    __syncthreads();

    v8f acc = {};
    const float* Ar = As + l16 * AS_STRIDE + 2 * half;  // A row M=l16, K base by half
    const v2f*   Bv = (const v2f*)(Wt + (size_t)(n0 + l16) * DIMF + 2 * half);

    #pragma unroll
    for (int ks = 0; ks < 32; ++ks) {
        const int k0 = ks * 4;
        v2f a, b;
        // A 16x4 fragment: v0 = K {k0 | k0+2} by half, v1 = K {k0+1 | k0+3}
        a.x = Ar[k0];
        a.y = Ar[k0 + 1];
        // B 4x16 fragment from transposed W: contiguous pair -> one b64 load
        b = Bv[ks * 2];
        acc = __builtin_amdgcn_wmma_f32_16x16x4_f32(false, a, false, b,
                                                    (short)0, acc, false, false);
    }

    float bias = cadd;
    if (bias1) bias += bias1[n0 + l16];
    if (bias2) bias += bias2[n0 + l16];

    // C/D layout: VGPR v -> M = v + 8*half, N = lane%16
    #pragma unroll
    for (int v = 0; v < 8; ++v) {
        const int row = m0 + v + 8 * half;
        C[(size_t)row * DIMF + n0 + l16] = acc[v] + bias;
    }
}

// ---------------------------------------------------------------------------
// PPI scatter: out[col] += xw[row] * dinv[row]*dinv[col]   (32 float4-chunks/edge)
// ---------------------------------------------------------------------------
__global__ __launch_bounds__(256) void ppi_scatter(
    const float* __restrict__ xw, const float* __restrict__ dinv,
    const int* __restrict__ row, const int* __restrict__ col,
    float* __restrict__ out)
{
    const int gid = blockIdx.x * blockDim.x + threadIdx.x;   // < EPPI*32
    const int e = gid >> 5;
    const int chunk = gid & 31;
    const int r = row[e];
    const int c = col[e];
    const float norm = dinv[r] * dinv[c];
    const float4 v = ((const float4*)(xw + (size_t)r * DIMF))[chunk];
    float* dst = out + (size_t)c * DIMF + chunk * 4;
    atomicAdd(dst + 0, v.x * norm);
    atomicAdd(dst + 1, v.y * norm);
    atomicAdd(dst + 2, v.z * norm);
    atomicAdd(dst + 3, v.w * norm);
}

// ---------------------------------------------------------------------------
// DTI scatter-mean: out[p] += msgs[d] / max(cnt[p],1)
// ---------------------------------------------------------------------------
__global__ __launch_bounds__(256) void dti_scatter(
    const float* __restrict__ msgs, const float* __restrict__ cnt,
    const int* __restrict__ drug, const int* __restrict__ prot,
    float* __restrict__ out)
{
    const int gid = blockIdx.x * blockDim.x + threadIdx.x;   // < EDTI*32
    const int e = gid >> 5;
    const int chunk = gid & 31;
    const int d = drug[e];
    const int p = prot[e];
    const float w = 1.0f / fmaxf(cnt[p], 1.0f);
    const float4 v = ((const float4*)(msgs + (size_t)d * DIMF))[chunk];
    float* dst = out + (size_t)p * DIMF + chunk * 4;
    atomicAdd(dst + 0, v.x * w);
    atomicAdd(dst + 1, v.y * w);
    atomicAdd(dst + 2, v.z * w);
    atomicAdd(dst + 3, v.w * w);
}

// ---------------------------------------------------------------------------
// Finalize: add GCN self-loop term, then LayerNorm. One wave32 per row.
// ---------------------------------------------------------------------------
__global__ __launch_bounds__(256) void finalize_ln(
    float* __restrict__ out, const float* __restrict__ xw,
    const float* __restrict__ dinv)
{
    const int lane = threadIdx.x & 31;
    const int row  = blockIdx.x * 8 + (threadIdx.x >> 5);   // NPROT % 8 == 0
    const float di = dinv[row];
    const float sn = di * di;                               // self-loop norm

    float4 y        = ((float4*)(out + (size_t)row * DIMF))[lane];
    const float4 xv = ((const float4*)(xw + (size_t)row * DIMF))[lane];
    y.x += xv.x * sn; y.y += xv.y * sn; y.z += xv.z * sn; y.w += xv.w * sn;

    float s = y.x + y.y + y.z + y.w;
    #pragma unroll
    for (int o = 16; o > 0; o >>= 1) s += __shfl_xor(s, o, 32);
    const float mu = s * (1.0f / 128.0f);

    const float d0 = y.x - mu, d1 = y.y - mu, d2 = y.z - mu, d3 = y.w - mu;
    float ss = d0 * d0 + d1 * d1 + d2 * d2 + d3 * d3;
    #pragma unroll
    for (int o = 16; o > 0; o >>= 1) ss += __shfl_xor(ss, o, 32);
    const float inv = rsqrtf(ss * (1.0f / 128.0f) + 1e-5f);

    y.x = d0 * inv; y.y = d1 * inv; y.z = d2 * inv; y.w = d3 * inv;
    ((float4*)(out + (size_t)row * DIMF))[lane] = y;
}

// ---------------------------------------------------------------------------
extern "C" void kernel_launch(void* const* d_in, const int* in_sizes, int n_in,
                              void* d_out, int out_size, void* d_ws, size_t ws_size,
                              hipStream_t stream) {
    const float* x_prot = (const float*)d_in[0];
    const float* x_drug = (const float*)d_in[1];
    const float* W_gcn  = (const float*)d_in[2];
    const float* b_gcn  = (const float*)d_in[3];
    const float* W_td   = (const float*)d_in[4];
    const float* b_td   = (const float*)d_in[5];
    const float* W_pr   = (const float*)d_in[6];
    const float* b_pr   = (const float*)d_in[7];
    const int*   ppi    = (const int*)d_in[8];   // [2, EPPI] flat: row then col
    const int*   dtid   = (const int*)d_in[9];   // dti_drug
    const int*   dtip   = (const int*)d_in[10];  // dti_prot
    float*       out    = (float*)d_out;

    // workspace carve: xw | msgs | deg(->dinv) | cnt | 3x Wt   (~62.5 MB)
    float* xw     = (float*)d_ws;
    float* msgs   = xw + (size_t)NPROT * DIMF;
    float* deg    = msgs + (size_t)NDRUG * DIMF;
    float* cnt    = deg + NPROT;
    float* Wt_gcn = cnt + NPROT;
    float* Wt_td  = Wt_gcn + DIMF * DIMF;
    float* Wt_pr  = Wt_td + DIMF * DIMF;

    const int* ppi_row = ppi;
    const int* ppi_col = ppi + EPPI;

    init_deg_cnt<<<(NPROT + 255) / 256, 256, 0, stream>>>(deg, cnt, NPROT);
    count_edges<<<(EPPI + 255) / 256, 256, 0, stream>>>(ppi_col, deg, EPPI);
    count_edges<<<(EDTI + 255) / 256, 256, 0, stream>>>(dtip, cnt, EDTI);
    to_dinv<<<(NPROT + 255) / 256, 256, 0, stream>>>(deg, NPROT);   // deg -> dinv

    transpose128<<<DIMF, DIMF, 0, stream>>>(W_gcn, Wt_gcn);
    transpose128<<<DIMF, DIMF, 0, stream>>>(W_td,  Wt_td);
    transpose128<<<DIMF, DIMF, 0, stream>>>(W_pr,  Wt_pr);

    // xw = x_prot @ W_gcn                  (b_gcn folded into accumulation below)
    gemm_wmma_f32<<<NPROT / 16, 256, 0, stream>>>(x_prot, Wt_gcn, nullptr, nullptr,
                                                  0.0f, xw);
    // msgs = x_drug @ W_td + b_td
    gemm_wmma_f32<<<NDRUG / 16, 256, 0, stream>>>(x_drug, Wt_td, b_td, nullptr,
                                                  0.0f, msgs);
    // out  = x_prot @ W_pr + b_pr + b_gcn + 1e-6  (accumulator for x1+x2+x3)
    gemm_wmma_f32<<<NPROT / 16, 256, 0, stream>>>(x_prot, Wt_pr, b_pr, b_gcn,
                                                  1e-6f, out);

    ppi_scatter<<<(EPPI * 32) / 256, 256, 0, stream>>>(xw, deg, ppi_row, ppi_col, out);
    dti_scatter<<<(EDTI * 32) / 256, 256, 0, stream>>>(msgs, cnt, dtid, dtip, out);

    finalize_ln<<<NPROT / 8, 256, 0, stream>>>(out, xw, deg);
}